// PredictDensityMatrix_24361054503167
// MI455X (gfx1250) — compile-verified
//
#include <hip/hip_runtime.h>

// CDNA5 (gfx1250) wave32: batched rho = T T^H / tr(.), one wave per 16x16
// complex matrix, 16 x v_wmma_f32_16x16x4_f32 per batch.
//
// T = A + iB (A = real part incl. diagonal, B = strictly-lower imag part).
//   Re(rho) = A A^T + B B^T      Im(rho) = B A^T - A B^T
// For the f32 16x16x4 shape, the A-slot operand (lane = row, k = v + 2*hi +
// 4*kc) and B-slot operand (lane = col, same k striping) use the identical
// per-lane gather, so one register set per matrix serves both slots.
// tr(T T^H) = ||T||_F^2 = sum of squares of ALL raw inputs (the /100 of the
// reference cancels in rho/tr, so it is skipped).

typedef __attribute__((ext_vector_type(2))) float v2f;
typedef __attribute__((ext_vector_type(4))) float v4f;
typedef __attribute__((ext_vector_type(8))) float v8f;

#define WAVES_PER_BLOCK 8

__global__ __launch_bounds__(256) void
PredictDensityMatrix_24361054503167_kernel(const float* __restrict__ in,
                                           float* __restrict__ out,
                                           int nbatch) {
    __shared__ float lds[WAVES_PER_BLOCK * 256];
    __shared__ float acc[WAVES_PER_BLOCK];

    const int tid   = threadIdx.x;
    const int wave  = tid >> 5;
    const int lane  = tid & 31;
    const int batch = blockIdx.x * WAVES_PER_BLOCK + wave;
    if (batch >= nbatch) return;          // no block-wide barriers below

    float* t    = &lds[wave * 256];
    float* pacc = &acc[wave];

    // --- Stage 1 KB coalesced: 2 x global_load_b128 -> 2 x ds_store_b128 ---
    const float* src = in + (size_t)batch * 256;
    const v4f x0 = *(const v4f*)(src + 4 * lane);
    const v4f x1 = *(const v4f*)(src + 128 + 4 * lane);
    *(v4f*)(t + 4 * lane)       = x0;
    *(v4f*)(t + 128 + 4 * lane) = x1;

    // --- Per-lane partial of tr = ||T||_F^2 over the 8 values this lane owns ---
    const float part = x0.x * x0.x + x0.y * x0.y + x0.z * x0.z + x0.w * x0.w
                     + x1.x * x1.x + x1.y * x1.y + x1.z * x1.z + x1.w * x1.w;

    // --- Wave reduction via one LDS float atomic (DS ops in-order per wave) ---
    *pacc = 0.0f;                 // all lanes write 0 to the same address
    atomicAdd(pacc, part);        // ds_add_f32, 32 lanes -> per-wave total

    // --- Branchless WMMA operand gather -----------------------------------
    const int hi = lane >> 4;            // half-wave selects K pair / row block
    const int m  = lane & 15;            // row (A-slot) / column (B-slot)
    const float diag = t[m];             // one ds_load_b32

    // float-index of off-diag pair: 16 + 2*(m(m-1)/2 + k), k = 4kc + v + 2hi
    //   = [16 + m(m-1) + 4hi] + (8kc + 2v)   -> per-slot immediates
    const int tb = 16 + m * (m - 1) + 4 * hi;

    v2f Aop[4], Bop[4];
#pragma unroll
    for (int kc = 0; kc < 4; ++kc) {
#pragma unroll
        for (int v = 0; v < 2; ++v) {
            const int  k  = 4 * kc + v + 2 * hi;
            const bool lt = (k < m);
            const bool eq = (k == m);
            // clamp to a safe aligned dummy inside this wave's region
            const int idx = lt ? (tb + 8 * kc + 2 * v) : 16;
            const v2f raw = *(const v2f*)(t + idx);      // ds_load_b64, aligned
            Aop[kc][v] = lt ? raw.x : (eq ? diag : 0.0f);
            Bop[kc][v] = lt ? raw.y : 0.0f;
        }
    }

    // --- 16 x v_wmma_f32_16x16x4_f32 --------------------------------------
    v8f re = {};
    v8f im = {};
#pragma unroll
    for (int kc = 0; kc < 4; ++kc)
        re = __builtin_amdgcn_wmma_f32_16x16x4_f32(false, Aop[kc], false, Aop[kc],
                                                   (short)0, re, false, false);
#pragma unroll
    for (int kc = 0; kc < 4; ++kc)
        re = __builtin_amdgcn_wmma_f32_16x16x4_f32(false, Bop[kc], false, Bop[kc],
                                                   (short)0, re, false, false);
#pragma unroll
    for (int kc = 0; kc < 4; ++kc)
        im = __builtin_amdgcn_wmma_f32_16x16x4_f32(false, Bop[kc], false, Aop[kc],
                                                   (short)0, im, false, false);
#pragma unroll
    for (int kc = 0; kc < 4; ++kc) {
        const v2f nA = -Aop[kc];     // f32 WMMA NEG bits are C-only: negate in regs
        im = __builtin_amdgcn_wmma_f32_16x16x4_f32(false, nA, false, Bop[kc],
                                                   (short)0, im, false, false);
    }

    // --- Normalize: tr read back (in-order after the atomic adds) ----------
    const float tr    = *pacc;
    const float scale = __builtin_amdgcn_rcpf(tr);

    // --- Store: C/D layout VGPR r -> (M = r + 8*hi, N = m); complex pairs.
    // One lane base + 8 b64 stores at immediate offsets 128*r bytes. ---
    float* dst = out + (size_t)batch * 512 + 2 * (128 * hi + m);
#pragma unroll
    for (int r = 0; r < 8; ++r) {
        v2f o;
        o.x = re[r] * scale;
        o.y = im[r] * scale;
        *(v2f*)(dst + 32 * r) = o;       // global_store_b64, coalesced
    }
}

extern "C" void kernel_launch(void* const* d_in, const int* in_sizes, int n_in,
                              void* d_out, int out_size, void* d_ws, size_t ws_size,
                              hipStream_t stream) {
    const float* in  = (const float*)d_in[0];
    float*       out = (float*)d_out;
    const int nbatch = in_sizes[0] / 256;                   // 65536
    const int blocks = (nbatch + WAVES_PER_BLOCK - 1) / WAVES_PER_BLOCK;
    PredictDensityMatrix_24361054503167_kernel<<<blocks, 256, 0, stream>>>(in, out, nbatch);
}